// GivensLinear_74835510165752
// MI455X (gfx1250) — compile-verified
//
#include <hip/hip_runtime.h>
#include <hip/hip_bf16.h>

#define DIM      128
#define NPAIRS   8128        // 128*127/2
#define RSTRIDE  129         // odd stride -> LDS bank-conflict-free row access
#define PSTRIDE  288         // floats per k-pair in swizzled W (256 data + 32 pad)
                             // 288 mod 64 == 32 banks -> lane halves use disjoint banks

typedef float v2f __attribute__((ext_vector_type(2)));
typedef float v8f __attribute__((ext_vector_type(8)));

// ---------------------------------------------------------------------------
// Kernel 1: build the 128x128 rotation matrix from 8128 sequential Givens
// rotations. One thread per ROW of W (rows evolve independently: step p only
// mixes W[r][i] and W[r][j]). Within an i-block, w_i is carried in a register
// so the serial dependency is a single FMA per step; w_j streams through LDS.
// Result is written to d_ws in the WMMA B-fragment swizzled layout:
//   element (k, n) -> wsw[(k>>1)*PSTRIDE + 2*n + (k&1)]
// so a lane's two K-values for one WMMA are one contiguous b64.
// ---------------------------------------------------------------------------
__global__ __launch_bounds__(128)
void build_w_kernel(const float* __restrict__ angles, float* __restrict__ wsw) {
    extern __shared__ float lds[];
    float* cs  = lds;                 // cs[2p]=cos, cs[2p+1]=sin  (16384 floats)
    float* row = lds + 16384;         // 128 rows, stride RSTRIDE
    const int tid = threadIdx.x;      // 0..127

    for (int p = tid; p < NPAIRS; p += 128) {
        float a = angles[p];
        cs[2 * p]     = cosf(a);
        cs[2 * p + 1] = sinf(a);
    }
    float* r = row + tid * RSTRIDE;
    #pragma unroll 4
    for (int c = 0; c < DIM; ++c) r[c] = (c == tid) ? 1.0f : 0.0f;
    __syncthreads();

    int p = 0;
    for (int i = 0; i < DIM - 1; ++i) {
        float wi = r[i];
        #pragma unroll 4
        for (int j = i + 1; j < DIM; ++j, ++p) {
            float c  = cs[2 * p];
            float s  = cs[2 * p + 1];
            float wj = r[j];
            r[j] = fmaf(wj, c, -(wi * s));   // off the serial chain
            wi   = fmaf(wi, c,  (wj * s));   // the only dependent op
        }
        r[i] = wi;
    }

    // thread tid holds W row k = tid; scatter into swizzled layout
    float* dst = wsw + (tid >> 1) * PSTRIDE + (tid & 1);
    #pragma unroll 4
    for (int n = 0; n < DIM; ++n) dst[2 * n] = r[n];
}

// ---------------------------------------------------------------------------
// Kernel 2: y = x @ W + bias via V_WMMA_F32_16X16X4_F32 (wave32).
// Block = 256 threads = 8 waves; each wave owns a 16x128 output tile
// (8 accumulators of v8f). W (swizzled) is staged in LDS once per block.
// B fragments are double-buffered in explicit registers so iteration t's
// WMMAs only depend on loads issued during iteration t-1 (no per-load stalls).
//
// A fragment (16x4 fp32): lane<16 : M=lane,   {K0,K1};
//                         lane>=16: M=lane-16,{K2,K3}   -> b64 from x.
// B fragment (4x16 fp32): lane<16 : N=lane,   {K0,K1};
//                         lane>=16: N=lane-16,{K2,K3}   -> b64 from LDS.
// C/D (16x16 fp32): VGPR r, lane l -> row (8*(l>=16)+r), col (l&15).
// ---------------------------------------------------------------------------
__global__ __launch_bounds__(256)
void givens_gemm_kernel(const float* __restrict__ x,
                        const float* __restrict__ wsw,
                        const float* __restrict__ bias,
                        float* __restrict__ out) {
    extern __shared__ float smem[];   // 64*PSTRIDE floats = 73728 B
    const int tid = threadIdx.x;

    {   // cooperative coalesced b128 copy of swizzled W into LDS
        const float4* src = (const float4*)wsw;
        float4*       dst = (float4*)smem;
        #pragma unroll
        for (int it = 0; it < (64 * PSTRIDE / 4) / 256; ++it)
            dst[it * 256 + tid] = src[it * 256 + tid];
    }

    const int wave = tid >> 5;
    const int lane = tid & 31;
    const int l15  = lane & 15;
    const int h    = lane >> 4;                 // which K-pair half
    const int m0   = blockIdx.x * 128 + wave * 16;

    const float* xsrc = x + (size_t)(m0 + l15) * DIM + 2 * h;
    const float* lw   = smem + h * PSTRIDE + 2 * l15;

    // bias values for this lane's 8 n-tiles: issue early, consumed in epilogue
    float bv[8];
    #pragma unroll
    for (int nt = 0; nt < 8; ++nt) bv[nt] = bias[nt * 16 + l15];

    v8f acc[8];
    #pragma unroll
    for (int nt = 0; nt < 8; ++nt)
        #pragma unroll
        for (int e = 0; e < 8; ++e) acc[nt][e] = 0.0f;

    __syncthreads();

    // ---- software-pipelined K loop: K = 128 in 32 steps of 4 ----
    v2f bbuf[2][8];
    v2f acur = *(const v2f*)(xsrc);
    #pragma unroll
    for (int nt = 0; nt < 8; ++nt)
        bbuf[0][nt] = *(const v2f*)(lw + nt * 32);

    #pragma unroll
    for (int t = 0; t < 32; ++t) {
        const int cb = t & 1;
        const int nb = cb ^ 1;
        v2f anext;
        if (t < 31) {
            anext = *(const v2f*)(xsrc + 4 * (t + 1));
            #pragma unroll
            for (int nt = 0; nt < 8; ++nt)
                bbuf[nb][nt] =
                    *(const v2f*)(lw + (t + 1) * (2 * PSTRIDE) + nt * 32);
        }
        #pragma unroll
        for (int nt = 0; nt < 8; ++nt) {
            acc[nt] = __builtin_amdgcn_wmma_f32_16x16x4_f32(
                false, acur, false, bbuf[cb][nt], (short)0, acc[nt],
                false, false);
        }
        if (t < 31) acur = anext;
    }

    // ---- epilogue: add bias, store ----
    float* obase = out + (size_t)(m0 + 8 * h) * DIM + l15;
    #pragma unroll
    for (int nt = 0; nt < 8; ++nt) {
        #pragma unroll
        for (int rr = 0; rr < 8; ++rr)
            obase[(size_t)rr * DIM + nt * 16] = acc[nt][rr] + bv[nt];
    }
}

// ---------------------------------------------------------------------------
extern "C" void kernel_launch(void* const* d_in, const int* in_sizes, int n_in,
                              void* d_out, int out_size, void* d_ws, size_t ws_size,
                              hipStream_t stream) {
    (void)n_in; (void)out_size; (void)ws_size;
    const float* x      = (const float*)d_in[0];   // (262144, 128) fp32
    const float* angles = (const float*)d_in[1];   // (8128,) fp32
    const float* bias   = (const float*)d_in[2];   // (128,) fp32
    float*       out    = (float*)d_out;
    float*       wsw    = (float*)d_ws;            // 64*PSTRIDE*4 = 73728 B

    const int rows    = in_sizes[0] / DIM;         // 262144
    const int nblocks = rows / 128;                // 2048

    build_w_kernel<<<1, 128, (16384 + 128 * RSTRIDE) * sizeof(float), stream>>>(
        angles, wsw);
    givens_gemm_kernel<<<nblocks, 256, 64 * PSTRIDE * sizeof(float), stream>>>(
        x, wsw, bias, out);
}